// CGEBlock_23072564314864
// MI455X (gfx1250) — compile-verified
//
#include <hip/hip_runtime.h>

typedef float v2f __attribute__((ext_vector_type(2)));
typedef float v8f __attribute__((ext_vector_type(8)));
typedef float f4  __attribute__((ext_vector_type(4)));

#define WAVES 2
#define BLOCK_THREADS (WAVES * 32)
#define ROWS_PER_WG (WAVES * 16)
#define XT_STRIDE 18    // [blade][row][m]: 16 m + pad2; banks ln*18%64 distinct
#define HS_STRIDE 34    // [blade][row][ch]: 32 ch + pad2; banks distinct
#define HP_STRIDE 260   // [row][ch*8+blade]: 256 + pad4; banks distinct
#define XT_FLOATS (8 * 16 * XT_STRIDE)   // 2304
#define HS_FLOATS (8 * 16 * HS_STRIDE)   // 4352
#define HP_FLOATS (16 * HP_STRIDE)       // 4160
#define WBUF_FLOATS (XT_FLOATS + HS_FLOATS + HP_FLOATS)  // 10816

namespace cge {
// Grade of each blade: (), e1,e2,e3, e12,e13,e23, e123
constexpr int GR[8] = {0, 1, 1, 1, 2, 2, 2, 3};
// Cayley product index: JT[i][k] = blade index of blade_i * blade_k
constexpr int JT[8][8] = {
    {0, 1, 2, 3, 4, 5, 6, 7},
    {1, 0, 4, 5, 2, 3, 7, 6},
    {2, 4, 0, 6, 1, 7, 3, 5},
    {3, 5, 6, 0, 7, 1, 2, 4},
    {4, 2, 1, 7, 0, 6, 5, 3},
    {5, 3, 7, 1, 6, 0, 4, 2},
    {6, 7, 3, 2, 5, 4, 0, 1},
    {7, 6, 5, 4, 3, 2, 1, 0}};
// Cayley sign: ST[i][k] = sign of blade_i * blade_k (Euclidean Cl(3,0))
constexpr float ST[8][8] = {
    { 1,  1,  1,  1,  1,  1,  1,  1},
    { 1,  1,  1,  1,  1,  1,  1,  1},
    { 1, -1,  1,  1, -1, -1,  1, -1},
    { 1, -1, -1,  1,  1, -1, -1,  1},
    { 1, -1,  1,  1, -1, -1,  1, -1},
    { 1, -1, -1,  1,  1, -1, -1,  1},
    { 1,  1, -1,  1, -1,  1, -1, -1},
    { 1,  1, -1,  1, -1,  1, -1, -1}};
constexpr int GB0[4] = {0, 1, 4, 7};  // first blade of grade
constexpr int GN[4]  = {1, 3, 3, 1};  // blades per grade
}  // namespace cge

__global__ __launch_bounds__(BLOCK_THREADS) void cge_fused_kernel(
    const float* __restrict__ x, const float* __restrict__ w1,
    const float* __restrict__ b1, const float* __restrict__ a_relu,
    const float* __restrict__ b_relu, const float* __restrict__ wl,
    const float* __restrict__ bl, const float* __restrict__ wr,
    const float* __restrict__ a_norm, const float* __restrict__ gp_w,
    const float* __restrict__ a_ln, float* __restrict__ out, int Bn) {
  using namespace cge;
  // ---- LDS (~140 KB total) ----
  __shared__ float w1s[4][32][18];   // [grade][n][m], B-fragment friendly
  __shared__ float wls[4][32][34];
  __shared__ float wrs[4][32][34];
  __shared__ float cfs[8][8][32];    // signed per-channel GP coefficients
  __shared__ float b1s[32], bls[32], alns[32];
  __shared__ float aRs[32][4], bRs[32][4], aNs[32][4];
  __shared__ float wbuf[WAVES][WBUF_FLOATS];  // per wave: xt + hs + hp

  const int tid = threadIdx.x;

  // ---- cooperative weight preprocessing into LDS ----
  for (int idx = tid; idx < 4 * 32 * 16; idx += BLOCK_THREADS) {
    int g = idx >> 9, n = (idx >> 4) & 31, m = idx & 15;
    w1s[g][n][m] = w1[n * 64 + m * 4 + g];
  }
  for (int idx = tid; idx < 4 * 32 * 32; idx += BLOCK_THREADS) {
    int g = idx >> 10, n = (idx >> 5) & 31, m = idx & 31;
    wls[g][n][m] = wl[n * 128 + m * 4 + g];
    wrs[g][n][m] = wr[n * 128 + m * 4 + g];
  }
  // GP coefficients: out_j += S[i][j] * gp_w[c,G[i],G[j],G[k]] * h_i * xr_k
  // with j = JT[i][k] (einsum contracts xr over Cayley's output axis).
  for (int idx = tid; idx < 8 * 8 * 32; idx += BLOCK_THREADS) {
    int i = idx >> 8, k = (idx >> 5) & 7, c = idx & 31;
    int j = JT[i][k];
    cfs[i][k][c] = ST[i][j] * gp_w[c * 64 + GR[i] * 16 + GR[j] * 4 + GR[k]];
  }
  if (tid < 32) {
    b1s[tid] = b1[tid];
    bls[tid] = bl[tid];
    alns[tid] = a_ln[tid];
#pragma unroll
    for (int g = 0; g < 4; ++g) {
      aRs[tid][g] = a_relu[tid * 4 + g];
      bRs[tid][g] = b_relu[tid * 4 + g];
      aNs[tid][g] = a_norm[tid * 4 + g];
    }
  }
  __syncthreads();

  const int wv = tid >> 5;
  const int lane = tid & 31;
  const int hi = lane >> 4;   // half-wave: selects K pair in frags, row+8 in D
  const int ln = lane & 15;
  const long b0 = (long)(blockIdx.x * WAVES + wv) * 16;
  if (b0 >= Bn) return;  // wave-uniform: EXEC stays all-ones for WMMA

  float* xt = &wbuf[wv][0];                        // [8 blade][16 row][18]
  float* hs = &wbuf[wv][XT_FLOATS];                // [8 blade][16 row][34]
  float* hp = &wbuf[wv][XT_FLOATS + HS_FLOATS];    // [16 row][256+4]

  // ---- stage x tile transposed to blade-major (A-fragments = 1x b64) ----
  {
    const float* xg = x + b0 * 128;
    const int m = lane >> 1;          // channel of this lane's float4
    const int i0 = (lane & 1) * 4;    // first blade of this lane's float4
#pragma unroll
    for (int q = 0; q < 16; ++q) {
      f4 v = *reinterpret_cast<const f4*>(&xg[q * 128 + lane * 4]);
      xt[((i0 + 0) * 16 + q) * XT_STRIDE + m] = v.x;
      xt[((i0 + 1) * 16 + q) * XT_STRIDE + m] = v.y;
      xt[((i0 + 2) * 16 + q) * XT_STRIDE + m] = v.z;
      xt[((i0 + 3) * 16 + q) * XT_STRIDE + m] = v.w;
    }
  }

  // ---- stage 1 (per N-half t): GEMM1 + MVReLU + spill (hp + hs) --------
  // h[b,n,i] = sum_m x[b,m,i] * w1[n,m,G[i]]   (v_wmma_f32_16x16x4_f32)
#pragma unroll
  for (int t = 0; t < 2; ++t) {
    v8f hA[8];
#pragma unroll
    for (int i = 0; i < 8; ++i) hA[i] = {};
#pragma unroll
    for (int g = 0; g < 4; ++g) {
#pragma unroll
      for (int kk = 0; kk < 4; ++kk) {
        const int m0 = kk * 4 + 2 * hi;  // K pair for this half-wave
        v2f b = *reinterpret_cast<const v2f*>(&w1s[g][t * 16 + ln][m0]);
#pragma unroll
        for (int u = 0; u < GN[g]; ++u) {
          const int i = GB0[g] + u;
          v2f a =
              *reinterpret_cast<const v2f*>(&xt[(i * 16 + ln) * XT_STRIDE + m0]);
          hA[i] = __builtin_amdgcn_wmma_f32_16x16x4_f32(
              false, a, false, b, (short)0, hA[i], false, false);
        }
      }
      __builtin_amdgcn_sched_barrier(0);
    }
    // MVReLU (lane-local; lane owns channel c for 8 batch rows)
    const int c = ln + 16 * t;
    const float bb = b1s[c];
    const float a0 = aRs[c][0], a1 = aRs[c][1], a2 = aRs[c][2], a3 = aRs[c][3];
    const float r0 = bRs[c][0], r1 = bRs[c][1], r2 = bRs[c][2], r3 = bRs[c][3];
#pragma unroll
    for (int r = 0; r < 8; ++r) {
      float h0 = hA[0][r] + bb;
      float h1 = hA[1][r], h2 = hA[2][r], h3 = hA[3][r];
      float h4 = hA[4][r], h5 = hA[5][r], h6 = hA[6][r];
      float h7 = hA[7][r];
      float i1 = h1 * h1 + h2 * h2 + h3 * h3;
      float i2 = h4 * h4 + h5 * h5 + h6 * h6;
      float i3 = h7 * h7;
      float g0 = fmaxf(0.0f, a0 * h0 + r0);
      float g1 = fmaxf(0.0f, a1 * i1 + r1);
      float g2 = fmaxf(0.0f, a2 * i2 + r2);
      float g3 = fmaxf(0.0f, a3 * i3 + r3);
      h0 *= g0; h1 *= g1; h2 *= g1; h3 *= g1;
      h4 *= g2; h5 *= g2; h6 *= g2; h7 *= g3;
      const int row = r + 8 * hi;
      // GP-friendly layout (b128 x2)
      f4 o0 = {h0, h1, h2, h3};
      f4 o1 = {h4, h5, h6, h7};
      *reinterpret_cast<f4*>(&hp[row * HP_STRIDE + c * 8]) = o0;
      *reinterpret_cast<f4*>(&hp[row * HP_STRIDE + c * 8 + 4]) = o1;
      // A-fragment layout (channel-contiguous per blade)
      hs[(0 * 16 + row) * HS_STRIDE + c] = h0;
      hs[(1 * 16 + row) * HS_STRIDE + c] = h1;
      hs[(2 * 16 + row) * HS_STRIDE + c] = h2;
      hs[(3 * 16 + row) * HS_STRIDE + c] = h3;
      hs[(4 * 16 + row) * HS_STRIDE + c] = h4;
      hs[(5 * 16 + row) * HS_STRIDE + c] = h5;
      hs[(6 * 16 + row) * HS_STRIDE + c] = h6;
      hs[(7 * 16 + row) * HS_STRIDE + c] = h7;
    }
    __builtin_amdgcn_sched_barrier(0);
  }

  // ---- stage 2 (per N-half t): fused wl/wr GEMMs + grade norm + GP -----
  v8f hl[8][2];
#pragma unroll
  for (int t = 0; t < 2; ++t) {
    const int c = ln + 16 * t;
#pragma unroll
    for (int i = 0; i < 8; ++i) hl[i][t] = {};
#pragma unroll
    for (int g = 0; g < 4; ++g) {
      v8f xr[3];
#pragma unroll
      for (int u = 0; u < GN[g]; ++u) xr[u] = {};
      // one A-fragment (b64) feeds both the wl-GEMM (hl) and wr-GEMM (xr)
#pragma unroll
      for (int kh = 0; kh < 2; ++kh) {
#pragma unroll
        for (int k2 = 0; k2 < 4; ++k2) {
          const int m0 = (kh * 4 + k2) * 4 + 2 * hi;
          v2f bL = *reinterpret_cast<const v2f*>(&wls[g][t * 16 + ln][m0]);
          v2f bR = *reinterpret_cast<const v2f*>(&wrs[g][t * 16 + ln][m0]);
#pragma unroll
          for (int u = 0; u < GN[g]; ++u) {
            const int i = GB0[g] + u;
            v2f a = *reinterpret_cast<const v2f*>(
                &hs[(i * 16 + ln) * HS_STRIDE + m0]);
            hl[i][t] = __builtin_amdgcn_wmma_f32_16x16x4_f32(
                false, a, false, bL, (short)0, hl[i][t], false, false);
            xr[u] = __builtin_amdgcn_wmma_f32_16x16x4_f32(
                false, a, false, bR, (short)0, xr[u], false, false);
          }
        }
        __builtin_amdgcn_sched_barrier(0);  // limit in-flight fragments
      }
      // grade norm: n = sigmoid(a_norm)*(||xr_g||-1)+1 ; xr /= (n+eps)
      const float sg = 1.0f / (1.0f + __expf(-aNs[c][g]));
#pragma unroll
      for (int r = 0; r < 8; ++r) {
        float nsq = 0.0f;
#pragma unroll
        for (int u = 0; u < GN[g]; ++u) {
          float v = xr[u][r];
          nsq += v * v;
        }
        float s = 1.0f / (sg * (sqrtf(nsq) - 1.0f) + 1.0f + 1e-6f);
#pragma unroll
        for (int u = 0; u < GN[g]; ++u) xr[u][r] *= s;
      }
      // GP: hl[JT[i][k]][t] += cfs[i][k][c] * h_i * xr_k (static j, unrolled)
      float cf[3][8];
#pragma unroll
      for (int u = 0; u < GN[g]; ++u) {
        const int kb = GB0[g] + u;
#pragma unroll
        for (int i = 0; i < 8; ++i) cf[u][i] = cfs[i][kb][c];
      }
#pragma unroll
      for (int r = 0; r < 8; ++r) {
        const int row = r + 8 * hi;
        f4 hv0 = *reinterpret_cast<const f4*>(&hp[row * HP_STRIDE + c * 8]);
        f4 hv1 = *reinterpret_cast<const f4*>(&hp[row * HP_STRIDE + c * 8 + 4]);
        const float hv[8] = {hv0.x, hv0.y, hv0.z, hv0.w,
                             hv1.x, hv1.y, hv1.z, hv1.w};
#pragma unroll
        for (int u = 0; u < GN[g]; ++u) {
          const int kb = GB0[g] + u;
          const float xv = xr[u][r];
#pragma unroll
          for (int i = 0; i < 8; ++i)
            hl[JT[i][kb]][t][r] += cf[u][i] * hv[i] * xv;
        }
      }
      __builtin_amdgcn_sched_barrier(0);
    }
  }

  // ---- bias(blade0) + 1/sqrt(2), MVLayerNorm butterfly, store ----------
  const float RS2 = 0.70710678118654752f;
#pragma unroll
  for (int t = 0; t < 2; ++t) {
    const float bb = bls[ln + 16 * t];
#pragma unroll
    for (int r = 0; r < 8; ++r) hl[0][t][r] += bb;
  }
#pragma unroll
  for (int i = 0; i < 8; ++i)
#pragma unroll
    for (int t = 0; t < 2; ++t)
#pragma unroll
      for (int r = 0; r < 8; ++r) hl[i][t][r] *= RS2;

#pragma unroll
  for (int r = 0; r < 8; ++r) {
    float s0 = 0.0f, s1 = 0.0f;
#pragma unroll
    for (int i = 0; i < 8; ++i) {
      s0 += hl[i][0][r] * hl[i][0][r];
      s1 += hl[i][1][r] * hl[i][1][r];
    }
    float v = sqrtf(s0) + sqrtf(s1);  // this lane's two channels
    v += __shfl_xor(v, 1);
    v += __shfl_xor(v, 2);
    v += __shfl_xor(v, 4);
    v += __shfl_xor(v, 8);            // sum over 32 ch (stays in 16-lane half)
    const float inv = 1.0f / (v * (1.0f / 32.0f) + 1e-6f);
    const int row = r + 8 * hi;
    float* op = out + (b0 + row) * 256;
#pragma unroll
    for (int t = 0; t < 2; ++t) {
      const int c = ln + 16 * t;
      const float sc = alns[c] * inv;
      f4 o0 = {hl[0][t][r] * sc, hl[1][t][r] * sc, hl[2][t][r] * sc,
               hl[3][t][r] * sc};
      f4 o1 = {hl[4][t][r] * sc, hl[5][t][r] * sc, hl[6][t][r] * sc,
               hl[7][t][r] * sc};
      *reinterpret_cast<f4*>(op + c * 8) = o0;
      *reinterpret_cast<f4*>(op + c * 8 + 4) = o1;
    }
  }
}

extern "C" void kernel_launch(void* const* d_in, const int* in_sizes, int n_in,
                              void* d_out, int out_size, void* d_ws,
                              size_t ws_size, hipStream_t stream) {
  const float* x      = (const float*)d_in[0];
  const float* w1     = (const float*)d_in[1];
  const float* b1     = (const float*)d_in[2];
  const float* a_relu = (const float*)d_in[3];
  const float* b_relu = (const float*)d_in[4];
  const float* wl     = (const float*)d_in[5];
  const float* bl     = (const float*)d_in[6];
  const float* wr     = (const float*)d_in[7];
  const float* a_norm = (const float*)d_in[8];
  const float* gp_w   = (const float*)d_in[9];
  const float* a_ln   = (const float*)d_in[10];
  // d_in[11] = cayley: structure hardcoded (deterministic Cl(3,0) table)

  const int Bn = in_sizes[0] / 128;  // batch rows (16 ch * 8 blades each)
  const int grid = (Bn + ROWS_PER_WG - 1) / ROWS_PER_WG;
  cge_fused_kernel<<<grid, BLOCK_THREADS, 0, stream>>>(
      x, w1, b1, a_relu, b_relu, wl, bl, wr, a_norm, gp_w, a_ln,
      (float*)d_out, Bn);
}